// GATLayer_72713796321965
// MI455X (gfx1250) — compile-verified
//
#include <hip/hip_runtime.h>

// GAT layer for MI455X (gfx1250, wave32).
//   inputs: x[50000,128] f32, edge_index[2,800000] int, W_fc[128,64] f32,
//           a_w[128,1] f32, a_b[1] f32
//   outputs (concat): out[50000,64] f32, alpha[800000] f32
//
// Phase 1: h_t = x @ W_fc via V_WMMA_F32_16X16X4_F32 (exact fp32).
// Phase 2: per-edge score -> exp -> segment-sum (hardware f32 atomics).
// Phase 3: alpha = h / h_sum[src]; out[src] += alpha * h_t[dst] (f32 atomics).

#define N_NODES  50000
#define N_FEAT   128
#define N_HID    64
#define N_EDGES  800000
#define LRELU_A  0.05f

typedef float v2f __attribute__((ext_vector_type(2)));
typedef float v8f __attribute__((ext_vector_type(8)));

// ---------------------------------------------------------------- zero fill
__global__ void zero_kernel(float* __restrict__ p, int n) {
    int i = blockIdx.x * blockDim.x + threadIdx.x;
    int stride = gridDim.x * blockDim.x;
    for (; i < n; i += stride) p[i] = 0.0f;
}

// ------------------------------------------------- Phase 1: h_t = x @ W_fc
// One wave computes a 16(M) x 64(N) strip: 4 fp32 WMMA accumulators,
// K=128 consumed 4 at a time -> 128 v_wmma_f32_16x16x4_f32 per wave.
// W_fc (32 KB) staged in LDS; x rows streamed from global (read once).
__global__ __launch_bounds__(256) void gemm_fc_wmma(
    const float* __restrict__ x, const float* __restrict__ W,
    float* __restrict__ h_t) {
    __shared__ float Wl[N_FEAT * N_HID];  // 32 KB of 320 KB/WGP

    const int tid = threadIdx.x;
    {   // cooperative 16B-vector load of W_fc into LDS
        const float4* Wv  = (const float4*)W;
        float4*       Wlv = (float4*)Wl;
        for (int i = tid; i < (N_FEAT * N_HID) / 4; i += 256) Wlv[i] = Wv[i];
    }
    __syncthreads();

    const int wave   = tid >> 5;
    const int lane   = tid & 31;
    const int m_tile = blockIdx.x * 8 + wave;
    if (m_tile >= N_NODES / 16) return;       // wave-uniform: EXEC stays all-1s

    const int m_base = m_tile * 16;
    const int row    = lane & 15;             // A: M index
    const int col    = lane & 15;             // B/C/D: N index within 16-tile
    const int kk     = (lane >> 4) << 1;      // lanes 16-31 hold K=2,3

    v8f acc0 = {}, acc1 = {}, acc2 = {}, acc3 = {};
    const float* xrow = x + (size_t)(m_base + row) * N_FEAT;

    for (int k = 0; k < N_FEAT; k += 4) {
        float2 av = *(const float2*)(xrow + k + kk);   // A frag (2 VGPRs)
        v2f a; a.x = av.x; a.y = av.y;
        const float* wr0 = Wl + (k + kk) * N_HID + col;
        const float* wr1 = wr0 + N_HID;
        v2f b0, b1, b2, b3;
        b0.x = wr0[0];  b0.y = wr1[0];
        b1.x = wr0[16]; b1.y = wr1[16];
        b2.x = wr0[32]; b2.y = wr1[32];
        b3.x = wr0[48]; b3.y = wr1[48];
        acc0 = __builtin_amdgcn_wmma_f32_16x16x4_f32(false, a, false, b0, (short)0, acc0, false, false);
        acc1 = __builtin_amdgcn_wmma_f32_16x16x4_f32(false, a, false, b1, (short)0, acc1, false, false);
        acc2 = __builtin_amdgcn_wmma_f32_16x16x4_f32(false, a, false, b2, (short)0, acc2, false, false);
        acc3 = __builtin_amdgcn_wmma_f32_16x16x4_f32(false, a, false, b3, (short)0, acc3, false, false);
    }

    // C/D layout: VGPR r -> M = r + 8*(lane>=16), N = lane&15
    const int hi = (lane >> 4) * 8;
    float* outrow = h_t + (size_t)(m_base + hi) * N_HID + col;
#pragma unroll
    for (int r = 0; r < 8; ++r) {
        float* p = outrow + (size_t)r * N_HID;
        p[0]  = acc0[r];
        p[16] = acc1[r];
        p[32] = acc2[r];
        p[48] = acc3[r];
    }
}

// --------------------------------- Phase 2: edge scores + segment softmax sum
// One edge per wave32: 32 lanes x float2 cover all 64 features of src and dst.
__global__ __launch_bounds__(256) void edge_score(
    const float* __restrict__ h_t, const int* __restrict__ ei,
    const float* __restrict__ a_w, const float* __restrict__ a_b,
    float* __restrict__ hbuf, float* __restrict__ h_sum) {
    const int lane = threadIdx.x & 31;
    const int wid  = (blockIdx.x * blockDim.x + threadIdx.x) >> 5;
    const int nw   = (gridDim.x * blockDim.x) >> 5;

    // attention vector halves, lane-resident for the whole loop
    const float a1x = a_w[2 * lane], a1y = a_w[2 * lane + 1];
    const float a2x = a_w[N_HID + 2 * lane], a2y = a_w[N_HID + 2 * lane + 1];
    const float bias = a_b[0];

    for (int e = wid; e < N_EDGES; e += nw) {
        const int s = ei[e];
        const int d = ei[N_EDGES + e];
        float2 p = *(const float2*)(h_t + (size_t)s * N_HID + 2 * lane);
        float2 q = *(const float2*)(h_t + (size_t)d * N_HID + 2 * lane);
        float sum = p.x * a1x + p.y * a1y + q.x * a2x + q.y * a2y;
#pragma unroll
        for (int off = 16; off > 0; off >>= 1)
            sum += __shfl_xor(sum, off, 32);
        float z = sum + bias;
        z = (z >= 0.0f) ? z : LRELU_A * z;     // LeakyReLU(0.05)
        float h = __expf(z);
        if (lane == 0) {
            hbuf[e] = h;
            unsafeAtomicAdd(h_sum + s, h);     // global_atomic_add_f32
        }
    }
}

// ------------------------- Phase 3: alpha + weighted scatter-add aggregation
__global__ __launch_bounds__(256) void edge_aggregate(
    const float* __restrict__ h_t, const int* __restrict__ ei,
    const float* __restrict__ hbuf, const float* __restrict__ h_sum,
    float* __restrict__ out, float* __restrict__ alpha_out) {
    const int lane = threadIdx.x & 31;
    const int wid  = (blockIdx.x * blockDim.x + threadIdx.x) >> 5;
    const int nw   = (gridDim.x * blockDim.x) >> 5;

    for (int e = wid; e < N_EDGES; e += nw) {
        const int s = ei[e];
        const int d = ei[N_EDGES + e];
        const float alpha = hbuf[e] / h_sum[s];
        if (lane == 0) alpha_out[e] = alpha;
        float2 q = *(const float2*)(h_t + (size_t)d * N_HID + 2 * lane);
        float* o = out + (size_t)s * N_HID + 2 * lane;
        unsafeAtomicAdd(o,     alpha * q.x);   // global_atomic_add_f32
        unsafeAtomicAdd(o + 1, alpha * q.y);
    }
}

// --------------------------------------------------------------------------
extern "C" void kernel_launch(void* const* d_in, const int* in_sizes, int n_in,
                              void* d_out, int out_size, void* d_ws, size_t ws_size,
                              hipStream_t stream) {
    (void)in_sizes; (void)n_in; (void)out_size; (void)ws_size;
    const float* x   = (const float*)d_in[0];
    const int*   ei  = (const int*)d_in[1];   // [2, N_EDGES]: src row then dst row
    const float* W   = (const float*)d_in[2]; // [128, 64]
    const float* a_w = (const float*)d_in[3]; // [128, 1]
    const float* a_b = (const float*)d_in[4]; // [1]

    float* out   = (float*)d_out;                  // [N_NODES * N_HID]
    float* alpha = out + (size_t)N_NODES * N_HID;  // [N_EDGES]

    float* h_t   = (float*)d_ws;                   // 12.8 MB, L2-resident
    float* hbuf  = h_t + (size_t)N_NODES * N_HID;  // 3.2 MB
    float* h_sum = hbuf + N_EDGES;                 // 0.2 MB

    // segment_sum targets start at zero every call (harness poisons buffers)
    zero_kernel<<<512, 256, 0, stream>>>(out, N_NODES * N_HID);
    zero_kernel<<<64, 256, 0, stream>>>(h_sum, N_NODES);

    // Phase 1: 3125 row-tiles of 16, 8 waves (tiles) per 256-thread block
    gemm_fc_wmma<<<(N_NODES / 16 + 7) / 8, 256, 0, stream>>>(x, W, h_t);

    // Phases 2-3: grid-stride, one edge per wave
    edge_score<<<1024, 256, 0, stream>>>(h_t, ei, a_w, a_b, hbuf, h_sum);
    edge_aggregate<<<1024, 256, 0, stream>>>(h_t, ei, hbuf, h_sum, out, alpha);
}